// GRU_90838558311148
// MI455X (gfx1250) — compile-verified
//
#include <hip/hip_runtime.h>
#include <hip/hip_bf16.h>
#include <stdint.h>

// ---------------------------------------------------------------------------
// GRU-style recurrence, T=512, B=64, D=H=1024, fp32 end-to-end.
// Strategy: single persistent kernel; per step each wave computes one 16x16
// (batch x hidden) tile of all 5 gate pre-activations as a fused K=2048 GEMM
// (K<1024: x_t @ W^T, K>=1024: h @ U^T) using V_WMMA_F32_16X16X4_F32, then the
// elementwise GRU update, keeping its h-tile in registers across timesteps.
// Grid-wide monotonic atomic barrier between steps. out[t] doubles as h_t.
// ---------------------------------------------------------------------------

typedef float v2f __attribute__((ext_vector_type(2)));
typedef float v4f __attribute__((ext_vector_type(4)));
typedef float v8f __attribute__((ext_vector_type(8)));

#define TT 512
#define BB 64
#define DD 1024
#define HH 1024
#define BH (BB * HH)

// one 16x32 f32 panel, padded to row stride 36 floats:
//  - 36 % 4 == 0 keeps b128 LDS stores 16B aligned
//  - b64 fragment reads hit distinct bank pairs across lanes
constexpr int PANEL = 16 * 36;

__device__ __forceinline__ v8f wmma4(v2f a, v2f b, v8f c) {
  // D = A(16x4,f32) * B(4x16,f32) + C(16x16,f32)
  return __builtin_amdgcn_wmma_f32_16x16x4_f32(false, a, false, b, (short)0, c,
                                               false, false);
}

__device__ __forceinline__ float sigmoidf_(float x) {
  return 1.0f / (1.0f + __expf(-x));
}

// Stage a 16-row x 32-col f32 panel (global row stride 1024) into LDS
// (row stride 36) using all 32 lanes with coalesced float4 loads.
__device__ __forceinline__ void stage_panel(float* __restrict__ dst,
                                            const float* __restrict__ src,
                                            int lane) {
  const int r0 = lane >> 3;        // 0..3
  const int c4 = (lane & 7) << 2;  // 0,4,..,28
#pragma unroll
  for (int p = 0; p < 4; ++p) {
    const int r = (p << 2) + r0;
    const v4f v = *(const v4f*)(src + r * 1024 + c4);
    *(v4f*)(dst + r * 36 + c4) = v;
  }
}

// One K=1024 half of the fused gate GEMM. asrc: 16 rows (stride 1024) of the
// A operand (x_t or h). wr[g]: 16 weight rows (stride 1024) for gate g.
__device__ __forceinline__ void gemm_half(const float* __restrict__ asrc,
                                          const float* const* __restrict__ wr,
                                          float* __restrict__ aP,
                                          float* __restrict__ bP, int lane,
                                          v8f& az, v8f& ar, v8f& as_, v8f& aq,
                                          v8f& ac) {
  // fragment offset: A lane holds A[m][k0 + 2*(l>>4) + {0,1}], m = l&15;
  // B lane holds    B[k0 + 2*(l>>4) + {0,1}][n],      n = l&15 (panel is [n][k])
  const int fo = (lane & 15) * 36 + ((lane >> 4) << 1);
  for (int kp = 0; kp < 32; ++kp) {
    const int kb = kp << 5;
    __builtin_prefetch(asrc + kb + 32 + (lane >> 3) * 1024 + ((lane & 7) << 2),
                       0, 1);  // global_prefetch_b8, next A panel
    stage_panel(aP, asrc + kb, lane);
    stage_panel(bP + 0 * PANEL, wr[0] + kb, lane);
    stage_panel(bP + 1 * PANEL, wr[1] + kb, lane);
    stage_panel(bP + 2 * PANEL, wr[2] + kb, lane);
    stage_panel(bP + 3 * PANEL, wr[3] + kb, lane);
    stage_panel(bP + 4 * PANEL, wr[4] + kb, lane);
#pragma unroll
    for (int ks = 0; ks < 8; ++ks) {
      const int k0 = (ks << 2) + fo;
      const v2f a = *(const v2f*)(aP + k0);
      az = wmma4(a, *(const v2f*)(bP + 0 * PANEL + k0), az);
      ar = wmma4(a, *(const v2f*)(bP + 1 * PANEL + k0), ar);
      as_ = wmma4(a, *(const v2f*)(bP + 2 * PANEL + k0), as_);
      aq = wmma4(a, *(const v2f*)(bP + 3 * PANEL + k0), aq);
      ac = wmma4(a, *(const v2f*)(bP + 4 * PANEL + k0), ac);
    }
  }
}

// ---------------------------------------------------------------------------
// Precompute kernels (time-invariant terms)
// ---------------------------------------------------------------------------
__global__ void esum_kernel(const float* __restrict__ et,
                            float* __restrict__ esum) {
  const int idx = blockIdx.x * blockDim.x + threadIdx.x;  // 65536
  const int b = idx >> 10, h = idx & 1023;
  float s = 0.0f;
#pragma unroll
  for (int k = 0; k < 16; ++k) s += et[((b << 4) + k) * 1024 + h];
  esum[idx] = s;
}

__global__ void proj_kernel(const float* __restrict__ g,
                            const float* __restrict__ yM,
                            const float* __restrict__ esum,
                            const float* __restrict__ zM,
                            const float* __restrict__ h0,
                            const float* __restrict__ ug,
                            float* __restrict__ gy, float* __restrict__ ez,
                            float* __restrict__ hinit) {
  const int idx = blockIdx.x * blockDim.x + threadIdx.x;  // 65536
  const int b = idx >> 10, h = idx & 1023;
  const float* grow = g + b * 1024;
  const float* yrow = yM + h * 1024;
  const float* erow = esum + b * 1024;
  const float* zrow = zM + h * 1024;
  const float* hrow = h0 + b * 1024;
  const float* urow = ug + h * 1024;
  float s1 = 0.0f, s2 = 0.0f, s3 = 0.0f;
  for (int d = 0; d < 1024; ++d) {
    s1 = fmaf(grow[d], yrow[d], s1);
    s2 = fmaf(erow[d], zrow[d], s2);
    s3 = fmaf(hrow[d], urow[d], s3);
  }
  gy[idx] = s1;
  ez[idx] = s2;
  hinit[idx] = s3;
}

// ---------------------------------------------------------------------------
// Persistent recurrent kernel: 64 blocks x 128 threads (256 waves).
// Wave job = (mt, nt): 16x16 output tile, mt in [0,4) batch tiles, nt in [0,64).
// ---------------------------------------------------------------------------
__global__ void __launch_bounds__(128) gru_persistent(
    const float* __restrict__ x,  // (T,B,D)
    const float* w_z, const float* w_r, const float* w_s, const float* w_q,
    const float* w_c, const float* u_z, const float* u_r, const float* u_s,
    const float* u_q, const float* u_c, const float* b_z, const float* b_r,
    const float* b_c, const float* __restrict__ gy, const float* __restrict__ ez,
    const float* __restrict__ hinit, float* __restrict__ out,
    float* __restrict__ hfin, unsigned* __restrict__ counter) {
  __shared__ __align__(16) float lds[4 * 6 * PANEL];  // 54 KB: 4 waves x (A + 5B)
  const int lane = threadIdx.x & 31;
  const int warp = threadIdx.x >> 5;
  const int job = blockIdx.x * 4 + warp;  // 0..255
  const int mt = job & 3;
  const int nt = job >> 2;

  float* wls = lds + warp * 6 * PANEL;
  float* aP = wls;
  float* bP = wls + PANEL;

  const int mrow0 = mt * 16;
  const int nrow0 = nt * 16;

  const float* wfirst[5] = {w_z + nrow0 * DD, w_r + nrow0 * DD, w_s + nrow0 * DD,
                            w_q + nrow0 * DD, w_c + nrow0 * DD};
  const float* wsec[5] = {u_z + nrow0 * HH, u_r + nrow0 * HH, u_s + nrow0 * HH,
                          u_q + nrow0 * HH, u_c + nrow0 * HH};

  // C/D layout: vgpr i, lanes 0-15 -> M=i, lanes 16-31 -> M=i+8; N = l&15
  const int hi = lane >> 4;
  const int hcol = nrow0 + (lane & 15);

  const float bz = b_z[hcol], br = b_r[hcol], bc = b_c[hcol];
  float gyv[8], ezv[8], hreg[8];
#pragma unroll
  for (int i = 0; i < 8; ++i) {
    const int brow = mrow0 + i + 8 * hi;
    gyv[i] = gy[brow * HH + hcol];
    ezv[i] = ez[brow * HH + hcol];
    hreg[i] = hinit[brow * HH + hcol];  // own h tile lives in registers
  }

  unsigned target = 0;
  for (int t = 0; t < TT; ++t) {
    const float* xt = x + (size_t)t * BB * DD + mrow0 * DD;
    const float* hp =
        (t == 0 ? hinit : out + (size_t)(t - 1) * BH) + mrow0 * HH;

    v8f az = {}, ar = {}, as_ = {}, aq = {}, acx = {}, ach = {};
    gemm_half(xt, wfirst, aP, bP, lane, az, ar, as_, aq, acx);  // x_t @ W^T
    gemm_half(hp, wsec, aP, bP, lane, az, ar, as_, aq, ach);    // h   @ U^T

    float* ot = out + (size_t)t * BH;
#pragma unroll
    for (int i = 0; i < 8; ++i) {
      const float zt = sigmoidf_(az[i] + bz);
      const float rt = sigmoidf_(ar[i] + br);
      const float st = sigmoidf_(as_[i]);
      const float qt = sigmoidf_(aq[i]);
      const float ct =
          tanhf(acx[i] + rt * ach[i] + st * gyv[i] + qt * ezv[i] + bc);
      const float hn = (1.0f - zt) * hreg[i] + zt * ct;
      hreg[i] = hn;
      const int brow = mrow0 + i + 8 * hi;
      ot[brow * HH + hcol] = hn;
      if (t == TT - 1) hfin[brow * HH + hcol] = hn;
    }

    // grid-wide barrier: h_t must be visible everywhere before step t+1
    target += gridDim.x;
    __threadfence();
    __syncthreads();
    if (threadIdx.x == 0) {
      __hip_atomic_fetch_add(counter, 1u, __ATOMIC_ACQ_REL,
                             __HIP_MEMORY_SCOPE_AGENT);
      while (__hip_atomic_load(counter, __ATOMIC_ACQUIRE,
                               __HIP_MEMORY_SCOPE_AGENT) < target) {
        __builtin_amdgcn_s_sleep(2);
      }
    }
    __syncthreads();
  }
}

// ---------------------------------------------------------------------------
extern "C" void kernel_launch(void* const* d_in, const int* in_sizes, int n_in,
                              void* d_out, int out_size, void* d_ws,
                              size_t ws_size, hipStream_t stream) {
  const float* x = (const float*)d_in[0];
  const float* g = (const float*)d_in[1];
  const float* et = (const float*)d_in[2];
  const float* h0 = (const float*)d_in[3];
  const float* w_z = (const float*)d_in[4];
  const float* u_z = (const float*)d_in[5];
  const float* b_z = (const float*)d_in[6];
  const float* w_r = (const float*)d_in[7];
  const float* u_r = (const float*)d_in[8];
  const float* b_r = (const float*)d_in[9];
  const float* w_c = (const float*)d_in[10];
  const float* u_c = (const float*)d_in[11];
  const float* b_c = (const float*)d_in[12];
  const float* zM = (const float*)d_in[13];
  const float* yM = (const float*)d_in[14];
  const float* w_s = (const float*)d_in[15];
  const float* u_s = (const float*)d_in[16];
  const float* w_q = (const float*)d_in[17];
  const float* u_q = (const float*)d_in[18];
  const float* u_g = (const float*)d_in[19];

  float* out = (float*)d_out;  // (T,B,H) then (B,H) h_final
  float* hfin = out + (size_t)TT * BH;

  char* wsb = (char*)d_ws;
  float* esum = (float*)(wsb);                       // (B,H)
  float* gy = (float*)(wsb + (size_t)BH * 4);        // (B,H)
  float* ez = (float*)(wsb + (size_t)BH * 8);        // (B,H)
  float* hinit = (float*)(wsb + (size_t)BH * 12);    // (B,H)
  unsigned* counter = (unsigned*)(wsb + (size_t)BH * 16);

  hipMemsetAsync(counter, 0, 256, stream);
  esum_kernel<<<BH / 256, 256, 0, stream>>>(et, esum);
  proj_kernel<<<BH / 256, 256, 0, stream>>>(g, yM, esum, zM, h0, u_g, gy, ez,
                                            hinit);
  gru_persistent<<<64, 128, 0, stream>>>(x, w_z, w_r, w_s, w_q, w_c, u_z, u_r,
                                         u_s, u_q, u_c, b_z, b_r, b_c, gy, ez,
                                         hinit, out, hfin, counter);
}